// DevEdge_24395414241576
// MI455X (gfx1250) — compile-verified
//
#include <hip/hip_runtime.h>
#include <math.h>

#define NN 50000
#define EE 400000
#define GG 256

typedef float v2f __attribute__((ext_vector_type(2)));
typedef float v8f __attribute__((ext_vector_type(8)));

__device__ __forceinline__ v8f wmma4(v2f a, v2f b, v8f c) {
  // V_WMMA_F32_16X16X4_F32 : D(16x16,f32) = A(16x4,f32) x B(4x16,f32) + C
  return __builtin_amdgcn_wmma_f32_16x16x4_f32(false, a, false, b, (short)0, c, false, false);
}

// ---------------------------------------------------------------- utilities
__global__ void fill_i32_kernel(int* __restrict__ p, int v, long long n) {
  long long i = (long long)blockIdx.x * blockDim.x + threadIdx.x;
  if (i < n) p[i] = v;
}
__global__ void fill_f32_kernel(float* __restrict__ p, float v, long long n) {
  long long i = (long long)blockIdx.x * blockDim.x + threadIdx.x;
  if (i < n) p[i] = v;
}

__device__ __forceinline__ int lower_bound_i(const int* a, int n, int key) {
  int lo = 0, hi = n;
  while (lo < hi) { int mid = (lo + hi) >> 1; if (a[mid] < key) lo = mid + 1; else hi = mid; }
  return lo;
}

// ---------------------------------------------------------------- edge features
__global__ void edge_feat_kernel(const float* __restrict__ x, const int* __restrict__ snd,
                                 const int* __restrict__ rcv, float* __restrict__ ef, int Ecnt) {
  int e = blockIdx.x * blockDim.x + threadIdx.x;
  if (e >= Ecnt) return;
  int s = snd[e], r = rcv[e];
  float d[6];
#pragma unroll
  for (int c = 0; c < 6; ++c) d[c] = x[(size_t)r * 6 + c] - x[(size_t)s * 6 + c];
  float dist = sqrtf(d[0] * d[0] + d[1] * d[1] + d[2] * d[2]);
  float inv = (dist > 0.f) ? 1.f / dist : 0.f;
  float* o = ef + (size_t)e * 8;
  o[0] = d[3]; o[1] = d[4]; o[2] = d[5]; o[3] = dist;
  o[4] = d[0] * inv; o[5] = d[1] * inv; o[6] = d[2] * inv; o[7] = 0.f;
}

// ---------------------------------------------------------------- edge MLP + segment-min (WMMA)
template<int DIN>
__device__ __forceinline__ float edge_feat_at(const float* hr, const float* hs, const float* ef, int k) {
  constexpr int DC = 2 * DIN + 7;
  if (k >= DC) return 0.f;
  if (k < DIN) return hr[k];
  if (k < 2 * DIN) return hs[k - DIN];
  return ef[k - 2 * DIN];
}

template<int DIN>
__global__ __launch_bounds__(64)
void edge_mlp_min_kernel(const float* __restrict__ H, const float* __restrict__ EF,
                         const int* __restrict__ rcv, const int* __restrict__ snd,
                         const float* __restrict__ W1, const float* __restrict__ B1,
                         const float* __restrict__ W2, const float* __restrict__ B2,
                         int* __restrict__ OutMin, int Ecnt) {
  constexpr int DC = 2 * DIN + 7;
  constexpr int KP = (DC + 3) & ~3;
  __shared__ float act[2][16][260];            // 33,280 B
  const int lane = threadIdx.x & 31;
  const int wave = threadIdx.x >> 5;
  const int mr   = lane & 15;                  // A-row / B,C column lane
  const int kh   = (lane >> 4) << 1;           // K pair selected by lane half
  const int mhi  = (lane >> 4) << 3;           // C row offset for lane half
  const int ebase = (blockIdx.x * 2 + wave) * 16;

  int ed = ebase + mr; if (ed >= Ecnt) ed = Ecnt - 1;   // clamp: min-scatter is idempotent
  const int rv = rcv[ed];
  const int sd = snd[ed];
  const float* hr = H + (size_t)rv * DIN;
  const float* hs = H + (size_t)sd * DIN;
  const float* ef = EF + (size_t)ed * 8;

  // ---- layer 1: [DC -> 256], relu, staged to LDS (two halves of 128 cols) ----
  for (int half = 0; half < 2; ++half) {
    v8f acc[8];
    for (int t = 0; t < 8; ++t) for (int j = 0; j < 8; ++j) acc[t][j] = 0.f;
    for (int k0 = 0; k0 < KP; k0 += 4) {
      const int ka = k0 + kh;
      v2f a;
      a.x = edge_feat_at<DIN>(hr, hs, ef, ka);
      a.y = edge_feat_at<DIN>(hr, hs, ef, ka + 1);
#pragma unroll
      for (int t = 0; t < 8; ++t) {
        const int n = half * 128 + t * 16 + mr;
        v2f b;
        b.x = (ka     < DC) ? W1[(size_t)ka       * 256 + n] : 0.f;
        b.y = (ka + 1 < DC) ? W1[(size_t)(ka + 1) * 256 + n] : 0.f;
        acc[t] = wmma4(a, b, acc[t]);
      }
    }
#pragma unroll
    for (int t = 0; t < 8; ++t) {
      const int n = half * 128 + t * 16 + mr;
      const float bias = B1[n];
#pragma unroll
      for (int r = 0; r < 8; ++r)
        act[wave][mhi + r][n] = fmaxf(acc[t][r] + bias, 0.f);
    }
  }
  __syncthreads();

  // ---- layer 2: [256 -> 128], relu, atomic-min scatter on float bits (>=0) ----
  v8f acc2[8];
  for (int t = 0; t < 8; ++t) for (int j = 0; j < 8; ++j) acc2[t][j] = 0.f;
  for (int k0 = 0; k0 < 256; k0 += 4) {
    const int ka = k0 + kh;
    v2f a;
    a.x = act[wave][mr][ka];
    a.y = act[wave][mr][ka + 1];
#pragma unroll
    for (int t = 0; t < 8; ++t) {
      const int n = t * 16 + mr;
      v2f b;
      b.x = W2[(size_t)ka       * 128 + n];
      b.y = W2[(size_t)(ka + 1) * 128 + n];
      acc2[t] = wmma4(a, b, acc2[t]);
    }
  }
  int nodeOf[8];
#pragma unroll
  for (int r = 0; r < 8; ++r) {
    int em = ebase + mhi + r; if (em >= Ecnt) em = Ecnt - 1;
    nodeOf[r] = rcv[em];
  }
#pragma unroll
  for (int t = 0; t < 8; ++t) {
    const int n = t * 16 + mr;
    const float bias = B2[n];
#pragma unroll
    for (int r = 0; r < 8; ++r) {
      const float v = fmaxf(acc2[t][r] + bias, 0.f);
      atomicMin(&OutMin[(size_t)nodeOf[r] * 128 + n], __float_as_int(v));
    }
  }
}

// ---------------------------------------------------------------- generic dense (WMMA)
template<int D0, int D1>
__device__ __forceinline__ float feat2(const float* x0, const float* x1, int k) {
  if (k >= D0 + D1) return 0.f;
  return (k < D0) ? x0[k] : x1[k - D0];
}

template<int D0, int D1, int ACT>   // ACT: 0 = none, 1 = relu
__global__ __launch_bounds__(256)
void dense_wmma_kernel(const float* __restrict__ X0, const float* __restrict__ X1,
                       const float* __restrict__ W, const float* __restrict__ Bv,
                       float* __restrict__ Y, int rows, int Dout) {
  constexpr int DIN = D0 + D1;
  constexpr int KP = (DIN + 3) & ~3;
  const int lane = threadIdx.x & 31;
  const int wave = threadIdx.x >> 5;
  const int mr  = lane & 15;
  const int kh  = (lane >> 4) << 1;
  const int mhi = (lane >> 4) << 3;
  const int row0 = (blockIdx.x * 8 + wave) * 16;
  if (row0 >= rows) return;
  const int nbase = blockIdx.y * 128;
  int ntiles = (Dout - nbase + 15) >> 4; if (ntiles > 8) ntiles = 8;

  int arow = row0 + mr; if (arow >= rows) arow = rows - 1;
  const float* x0p = X0 + (size_t)arow * D0;
  const float* x1p = (D1 > 0) ? (X1 + (size_t)arow * D1) : X0;

  v8f acc[8];
  for (int t = 0; t < 8; ++t) for (int j = 0; j < 8; ++j) acc[t][j] = 0.f;
  for (int k0 = 0; k0 < KP; k0 += 4) {
    const int ka = k0 + kh;
    v2f a;
    a.x = feat2<D0, D1>(x0p, x1p, ka);
    a.y = feat2<D0, D1>(x0p, x1p, ka + 1);
    for (int t = 0; t < ntiles; ++t) {
      const int n = nbase + t * 16 + mr;
      v2f b;
      b.x = (ka     < DIN) ? W[(size_t)ka       * Dout + n] : 0.f;
      b.y = (ka + 1 < DIN) ? W[(size_t)(ka + 1) * Dout + n] : 0.f;
      acc[t] = wmma4(a, b, acc[t]);
    }
  }
  for (int t = 0; t < ntiles; ++t) {
    const int n = nbase + t * 16 + mr;
    const float bias = Bv[n];
    for (int r = 0; r < 8; ++r) {
      const int m = row0 + mhi + r;
      if (m < rows) {
        float v = acc[t][r] + bias;
        if (ACT == 1) v = fmaxf(v, 0.f);
        Y[(size_t)m * Dout + n] = v;
      }
    }
  }
}

// ---------------------------------------------------------------- GraphSage aggregation
__global__ void deg_kernel(const int* __restrict__ rcv, int* __restrict__ deg, int Ecnt) {
  int e = blockIdx.x * blockDim.x + threadIdx.x;
  if (e < Ecnt) atomicAdd(&deg[rcv[e]], 1);
}
__global__ void agg_kernel(const float* __restrict__ Hsrc, const int* __restrict__ rcv,
                           const int* __restrict__ snd, float* __restrict__ Agg, int Ecnt, int D) {
  long long i = (long long)blockIdx.x * blockDim.x + threadIdx.x;
  if (i >= (long long)Ecnt * D) return;
  int e = (int)(i / D), c = (int)(i % D);
  atomicAdd(&Agg[(size_t)rcv[e] * D + c], Hsrc[(size_t)snd[e] * D + c]);
}
__global__ void scale_kernel(float* __restrict__ Agg, const int* __restrict__ deg, int Nn, int D) {
  long long i = (long long)blockIdx.x * blockDim.x + threadIdx.x;
  if (i >= (long long)Nn * D) return;
  Agg[i] /= (float)deg[i / D];
}
__global__ void l2norm_relu_kernel(float* __restrict__ Hm, int rows, int D) {
  int gw = (blockIdx.x * blockDim.x + threadIdx.x) >> 5;
  int lane = threadIdx.x & 31;
  if (gw >= rows) return;
  float* row = Hm + (size_t)gw * D;
  float ss = 0.f;
  for (int c = lane; c < D; c += 32) { float v = row[c]; ss += v * v; }
  for (int off = 16; off > 0; off >>= 1) ss += __shfl_xor(ss, off, 32);
  float s = rsqrtf(fmaxf(ss, 1e-12f));
  for (int c = lane; c < D; c += 32) row[c] = fmaxf(row[c] * s, 0.f);
}

// ---------------------------------------------------------------- per-graph stats & pools
__global__ void graph_stats_kernel(const float* __restrict__ x, const int* __restrict__ gi,
                                   float* __restrict__ z, int Nn) {
  const int g = blockIdx.x;
  const int lo = lower_bound_i(gi, Nn, g);
  const int hi = lower_bound_i(gi, Nn, g + 1);
  const int c = threadIdx.x;
  if (c >= 6) return;
  float s = 0.f, s2 = 0.f, mx = -3.4e38f, mn = 3.4e38f;
  for (int nrow = lo; nrow < hi; ++nrow) {
    float v = x[(size_t)nrow * 6 + c];
    s += v; s2 += v * v; mx = fmaxf(mx, v); mn = fminf(mn, v);
  }
  float cnt = (float)(hi - lo);
  float avg = s / cnt;
  float var = fabsf(s2 / cnt - avg * avg);
  float* zr = z + (size_t)g * 1560 + 1536;
  zr[c] = avg; zr[6 + c] = var; zr[12 + c] = mx; zr[18 + c] = mn;
}
__global__ void graph_pool_kernel(const float* __restrict__ Hm, const int* __restrict__ gi,
                                  float* __restrict__ z, int Nn) {
  const int g = blockIdx.x;
  const int lo = lower_bound_i(gi, Nn, g);
  const int hi = lower_bound_i(gi, Nn, g + 1);
  const float inv = 1.f / (float)(hi - lo);
  for (int c = threadIdx.x; c < 512; c += blockDim.x) {
    float mx = -3.4e38f, s = 0.f;
    for (int nrow = lo; nrow < hi; ++nrow) {
      float v = Hm[(size_t)nrow * 512 + c];
      mx = fmaxf(mx, v); s += v;
    }
    float* zr = z + (size_t)g * 1560;
    zr[c] = mx; zr[512 + c] = s * inv; zr[1024 + c] = s;
  }
}

// ---------------------------------------------------------------- decoder pieces
__global__ void leaky_bn_kernel(float* __restrict__ Zb, const float* __restrict__ ga,
                                const float* __restrict__ be, const float* __restrict__ me,
                                const float* __restrict__ va, int total, int D) {
  int i = blockIdx.x * blockDim.x + threadIdx.x;
  if (i >= total) return;
  int c = i % D;
  float x = Zb[i];
  x = (x >= 0.f) ? x : 0.15f * x;
  Zb[i] = (x - me[c]) * rsqrtf(va[c] + 1e-3f) * ga[c] + be[c];
}
__global__ void dense_small_kernel(const float* __restrict__ X, const float* __restrict__ W,
                                   const float* __restrict__ Bv, float* __restrict__ Y,
                                   int Din, int Dout, int act) {  // act: 0 none, 1 sigmoid, 2 abs+eps
  const int r = blockIdx.x;
  const int c = threadIdx.x;
  if (c >= Dout) return;
  float s = Bv[c];
  const float* xr = X + (size_t)r * Din;
  for (int k = 0; k < Din; ++k) s += xr[k] * W[(size_t)k * Dout + c];
  if (act == 1) s = 1.f / (1.f + expf(-s));
  else if (act == 2) s = fabsf(s) + 1e-5f;
  Y[(size_t)r * Dout + c] = s;
}
__global__ void finalize_kernel(const float* __restrict__ lg, const float* __restrict__ zen,
                                const float* __restrict__ sg, float* __restrict__ out, int Gn) {
  int g = blockIdx.x * blockDim.x + threadIdx.x;
  if (g >= Gn) return;
  out[g * 5 + 0] = lg[g];
  out[g * 5 + 1] = zen[g * 2 + 0] * 3.14159265358979323846f;
  out[g * 5 + 2] = zen[g * 2 + 1] * 6.28318530717958647692f;
  out[g * 5 + 3] = sg[g * 2 + 0];
  out[g * 5 + 4] = sg[g * 2 + 1];
}

// ---------------------------------------------------------------- host side
struct NetParams {
  const float* x; const int* send; const int* recv; const int* gi;
  const float *sg1m_W0,*sg1m_b0,*sg1m_W1,*sg1m_b1;
  const float *sg1u_W0,*sg1u_b0,*sg1u_W1,*sg1u_b1;
  const float *sg2m0_W0,*sg2m0_b0,*sg2m0_W1,*sg2m0_b1;
  const float *sg2m1_W0,*sg2m1_b0,*sg2m1_W1,*sg2m1_b1;
  const float *sg2u_W0,*sg2u_b0,*sg2u_W1,*sg2u_b1;
  const float *gcn1_W,*gcn1_b,*gcn2_W,*gcn2_b;
  const float *dec1_W,*dec1_b,*dec2_W,*dec2_b;
  const float *bn1_g,*bn1_b,*bn1_m,*bn1_v;
  const float *bn2_g,*bn2_b,*bn2_m,*bn2_v;
  const float *loge_W0,*loge_b0,*loge_W1,*loge_b1,*loge_W2,*loge_b2;
  const float *ang_W0,*ang_b0,*ang_W1,*ang_b1,*ang_W2,*ang_b2;
  const float *as_W,*as_b;
  const float *sig_W0,*sig_b0,*sig_W1,*sig_b1,*sig_W2,*sig_b2;
};

static inline const float* Fp(void* const* d, int i) { return (const float*)d[i]; }
static inline const int*   Ip(void* const* d, int i) { return (const int*)d[i]; }

static void mapInputs(void* const* d, const int* sz, int n, NetParams& P) {
  (void)n;
  if (sz[0] == NN * 6) {
    // insertion order: x, send, receive, i, params{insertion}
    int k = 0;
    P.x = Fp(d,k++); P.send = Ip(d,k++); P.recv = Ip(d,k++); P.gi = Ip(d,k++);
    P.sg1m_W0=Fp(d,k++); P.sg1m_b0=Fp(d,k++); P.sg1m_W1=Fp(d,k++); P.sg1m_b1=Fp(d,k++);
    P.sg1u_W0=Fp(d,k++); P.sg1u_b0=Fp(d,k++); P.sg1u_W1=Fp(d,k++); P.sg1u_b1=Fp(d,k++);
    P.sg2m0_W0=Fp(d,k++); P.sg2m0_b0=Fp(d,k++); P.sg2m0_W1=Fp(d,k++); P.sg2m0_b1=Fp(d,k++);
    P.sg2m1_W0=Fp(d,k++); P.sg2m1_b0=Fp(d,k++); P.sg2m1_W1=Fp(d,k++); P.sg2m1_b1=Fp(d,k++);
    P.sg2u_W0=Fp(d,k++); P.sg2u_b0=Fp(d,k++); P.sg2u_W1=Fp(d,k++); P.sg2u_b1=Fp(d,k++);
    P.gcn1_W=Fp(d,k++); P.gcn1_b=Fp(d,k++); P.gcn2_W=Fp(d,k++); P.gcn2_b=Fp(d,k++);
    P.dec1_W=Fp(d,k++); P.dec1_b=Fp(d,k++); P.dec2_W=Fp(d,k++); P.dec2_b=Fp(d,k++);
    P.bn1_g=Fp(d,k++); P.bn1_b=Fp(d,k++); P.bn1_m=Fp(d,k++); P.bn1_v=Fp(d,k++);
    P.bn2_g=Fp(d,k++); P.bn2_b=Fp(d,k++); P.bn2_m=Fp(d,k++); P.bn2_v=Fp(d,k++);
    P.loge_W0=Fp(d,k++); P.loge_b0=Fp(d,k++); P.loge_W1=Fp(d,k++); P.loge_b1=Fp(d,k++);
    P.loge_W2=Fp(d,k++); P.loge_b2=Fp(d,k++);
    P.ang_W0=Fp(d,k++); P.ang_b0=Fp(d,k++); P.ang_W1=Fp(d,k++); P.ang_b1=Fp(d,k++);
    P.ang_W2=Fp(d,k++); P.ang_b2=Fp(d,k++);
    P.as_W=Fp(d,k++); P.as_b=Fp(d,k++);
    P.sig_W0=Fp(d,k++); P.sig_b0=Fp(d,k++); P.sig_W1=Fp(d,k++); P.sig_b1=Fp(d,k++);
    P.sig_W2=Fp(d,k++); P.sig_b2=Fp(d,k++);
  } else {
    // jax tree-flatten sorted-key order: i, params{sorted}, receive, send, x
    int k = 0;
    P.gi = Ip(d,k++);
    P.as_W=Fp(d,k++); P.as_b=Fp(d,k++);
    P.ang_W0=Fp(d,k++); P.ang_b0=Fp(d,k++); P.ang_W1=Fp(d,k++); P.ang_b1=Fp(d,k++);
    P.ang_W2=Fp(d,k++); P.ang_b2=Fp(d,k++);
    P.bn1_b=Fp(d,k++); P.bn1_g=Fp(d,k++); P.bn1_m=Fp(d,k++); P.bn1_v=Fp(d,k++);
    P.bn2_b=Fp(d,k++); P.bn2_g=Fp(d,k++); P.bn2_m=Fp(d,k++); P.bn2_v=Fp(d,k++);
    P.dec1_W=Fp(d,k++); P.dec1_b=Fp(d,k++); P.dec2_W=Fp(d,k++); P.dec2_b=Fp(d,k++);
    P.gcn1_W=Fp(d,k++); P.gcn1_b=Fp(d,k++); P.gcn2_W=Fp(d,k++); P.gcn2_b=Fp(d,k++);
    P.loge_W0=Fp(d,k++); P.loge_b0=Fp(d,k++); P.loge_W1=Fp(d,k++); P.loge_b1=Fp(d,k++);
    P.loge_W2=Fp(d,k++); P.loge_b2=Fp(d,k++);
    P.sg1m_W0=Fp(d,k++); P.sg1m_b0=Fp(d,k++); P.sg1m_W1=Fp(d,k++); P.sg1m_b1=Fp(d,k++);
    P.sg1u_W0=Fp(d,k++); P.sg1u_b0=Fp(d,k++); P.sg1u_W1=Fp(d,k++); P.sg1u_b1=Fp(d,k++);
    P.sg2m0_W0=Fp(d,k++); P.sg2m0_b0=Fp(d,k++); P.sg2m0_W1=Fp(d,k++); P.sg2m0_b1=Fp(d,k++);
    P.sg2m1_W0=Fp(d,k++); P.sg2m1_b0=Fp(d,k++); P.sg2m1_W1=Fp(d,k++); P.sg2m1_b1=Fp(d,k++);
    P.sg2u_W0=Fp(d,k++); P.sg2u_b0=Fp(d,k++); P.sg2u_W1=Fp(d,k++); P.sg2u_b1=Fp(d,k++);
    P.sig_W0=Fp(d,k++); P.sig_b0=Fp(d,k++); P.sig_W1=Fp(d,k++); P.sig_b1=Fp(d,k++);
    P.sig_W2=Fp(d,k++); P.sig_b2=Fp(d,k++);
    P.recv = Ip(d,k++); P.send = Ip(d,k++); P.x = Fp(d,k++);
  }
}

static inline unsigned cdivu(long long a, long long b) { return (unsigned)((a + b - 1) / b); }

extern "C" void kernel_launch(void* const* d_in, const int* in_sizes, int n_in,
                              void* d_out, int out_size, void* d_ws, size_t ws_size,
                              hipStream_t stream) {
  (void)out_size; (void)ws_size;
  NetParams P;
  mapInputs(d_in, in_sizes, n_in, P);

  char* ws = (char*)d_ws;
  size_t off = 0;
  auto alloc = [&](size_t bytes) -> char* {
    char* p = ws + off; off = (off + bytes + 255) & ~(size_t)255; return p;
  };
  float* efeat = (float*)alloc((size_t)EE * 8 * 4);
  float* bufA  = (float*)alloc((size_t)NN * 128 * 4);   // contiguous with bufB
  float* bufB  = (float*)alloc((size_t)NN * 128 * 4);
  float* bufC  = (float*)alloc((size_t)NN * 64 * 4);
  float* bufD  = (float*)alloc((size_t)NN * 256 * 4);
  float* bufE  = (float*)alloc((size_t)NN * 512 * 4);
  int*   deg   = (int*)  alloc((size_t)NN * 4);
  float* z     = (float*)alloc((size_t)GG * 1560 * 4);
  float* zb1   = (float*)alloc((size_t)GG * 1024 * 4);
  float* zb2   = (float*)alloc((size_t)GG * 512 * 4);
  float* hbA   = (float*)alloc((size_t)GG * 64 * 4);
  float* hbB   = (float*)alloc((size_t)GG * 64 * 4);
  float* oLog  = (float*)alloc((size_t)GG * 4);
  float* oAng  = (float*)alloc((size_t)GG * 2 * 4);
  float* oZen  = (float*)alloc((size_t)GG * 2 * 4);
  float* oSig  = (float*)alloc((size_t)GG * 2 * 4);

  const int FMAXBITS = 0x7f7fffff;
  const unsigned gN = cdivu(cdivu(NN, 16), 8);   // dense grid.x for N rows
  const unsigned gG = cdivu(cdivu(GG, 16), 8);   // dense grid.x for G rows
  const unsigned gE32 = cdivu(EE, 32);

  // edge features + raw-feature graph stats (fills z cols 1536..1559)
  edge_feat_kernel<<<cdivu(EE,256), 256, 0, stream>>>(P.x, P.send, P.recv, efeat, EE);
  graph_stats_kernel<<<GG, 32, 0, stream>>>(P.x, P.gi, z, NN);

  // ---- SG block 1: hop(x) -> min, then update MLP ----
  fill_i32_kernel<<<cdivu((long long)NN*128,256), 256, 0, stream>>>((int*)bufA, FMAXBITS, (long long)NN*128);
  edge_mlp_min_kernel<6><<<gE32, 64, 0, stream>>>(P.x, efeat, P.recv, P.send,
      P.sg1m_W0, P.sg1m_b0, P.sg1m_W1, P.sg1m_b1, (int*)bufA, EE);
  dense_wmma_kernel<128,0,1><<<dim3(gN,1), 256, 0, stream>>>(bufA, bufA, P.sg1u_W0, P.sg1u_b0, bufB, NN, 128);
  dense_wmma_kernel<128,0,1><<<dim3(gN,1), 256, 0, stream>>>(bufB, bufB, P.sg1u_W1, P.sg1u_b1, bufC, NN, 64);

  // ---- SG block 2: two hops, then update MLP ----
  fill_i32_kernel<<<cdivu((long long)NN*128,256), 256, 0, stream>>>((int*)bufB, FMAXBITS, (long long)NN*128);
  edge_mlp_min_kernel<64><<<gE32, 64, 0, stream>>>(bufC, efeat, P.recv, P.send,
      P.sg2m0_W0, P.sg2m0_b0, P.sg2m0_W1, P.sg2m0_b1, (int*)bufB, EE);
  fill_i32_kernel<<<cdivu((long long)NN*128,256), 256, 0, stream>>>((int*)bufA, FMAXBITS, (long long)NN*128);
  edge_mlp_min_kernel<128><<<gE32, 64, 0, stream>>>(bufB, efeat, P.recv, P.send,
      P.sg2m1_W0, P.sg2m1_b0, P.sg2m1_W1, P.sg2m1_b1, (int*)bufA, EE);
  dense_wmma_kernel<128,0,1><<<dim3(gN,1), 256, 0, stream>>>(bufA, bufA, P.sg2u_W0, P.sg2u_b0, bufB, NN, 128);
  dense_wmma_kernel<128,0,1><<<dim3(gN,1), 256, 0, stream>>>(bufB, bufB, P.sg2u_W1, P.sg2u_b1, bufC, NN, 64);

  // ---- GraphSage convs ----
  fill_i32_kernel<<<cdivu(NN,256), 256, 0, stream>>>(deg, 0, NN);
  deg_kernel<<<cdivu(EE,256), 256, 0, stream>>>(P.recv, deg, EE);

  fill_f32_kernel<<<cdivu((long long)NN*64,256), 256, 0, stream>>>(bufA, 0.f, (long long)NN*64);
  agg_kernel<<<cdivu((long long)EE*64,256), 256, 0, stream>>>(bufC, P.recv, P.send, bufA, EE, 64);
  scale_kernel<<<cdivu((long long)NN*64,256), 256, 0, stream>>>(bufA, deg, NN, 64);
  dense_wmma_kernel<64,64,0><<<dim3(gN,2), 256, 0, stream>>>(bufC, bufA, P.gcn1_W, P.gcn1_b, bufD, NN, 256);
  l2norm_relu_kernel<<<cdivu((long long)NN*32,256), 256, 0, stream>>>(bufD, NN, 256);

  fill_f32_kernel<<<cdivu((long long)NN*256,256), 256, 0, stream>>>(bufA, 0.f, (long long)NN*256);
  agg_kernel<<<cdivu((long long)EE*256,256), 256, 0, stream>>>(bufD, P.recv, P.send, bufA, EE, 256);
  scale_kernel<<<cdivu((long long)NN*256,256), 256, 0, stream>>>(bufA, deg, NN, 256);
  dense_wmma_kernel<256,256,0><<<dim3(gN,4), 256, 0, stream>>>(bufD, bufA, P.gcn2_W, P.gcn2_b, bufE, NN, 512);
  l2norm_relu_kernel<<<cdivu((long long)NN*32,256), 256, 0, stream>>>(bufE, NN, 512);

  // ---- global pools into z (cols 0..1535) ----
  graph_pool_kernel<<<GG, 256, 0, stream>>>(bufE, P.gi, z, NN);

  // ---- decoder ----
  dense_wmma_kernel<1560,0,0><<<dim3(gG,8), 256, 0, stream>>>(z, z, P.dec1_W, P.dec1_b, zb1, GG, 1024);
  leaky_bn_kernel<<<cdivu((long long)GG*1024,256), 256, 0, stream>>>(zb1, P.bn1_g, P.bn1_b, P.bn1_m, P.bn1_v, GG*1024, 1024);
  dense_wmma_kernel<1024,0,0><<<dim3(gG,4), 256, 0, stream>>>(zb1, zb1, P.dec2_W, P.dec2_b, zb2, GG, 512);
  leaky_bn_kernel<<<cdivu((long long)GG*512,256), 256, 0, stream>>>(zb2, P.bn2_g, P.bn2_b, P.bn2_m, P.bn2_v, GG*512, 512);

  // ---- heads ----
  dense_small_kernel<<<GG, 64, 0, stream>>>(zb2, P.loge_W0, P.loge_b0, hbA, 512, 64, 0);
  dense_small_kernel<<<GG, 64, 0, stream>>>(hbA, P.loge_W1, P.loge_b1, hbB, 64, 64, 0);
  dense_small_kernel<<<GG, 64, 0, stream>>>(hbB, P.loge_W2, P.loge_b2, oLog, 64, 1, 0);

  dense_small_kernel<<<GG, 64, 0, stream>>>(zb2, P.ang_W0, P.ang_b0, hbA, 512, 64, 0);
  dense_small_kernel<<<GG, 64, 0, stream>>>(hbA, P.ang_W1, P.ang_b1, hbB, 64, 64, 0);
  dense_small_kernel<<<GG, 64, 0, stream>>>(hbB, P.ang_W2, P.ang_b2, oAng, 64, 2, 0);
  dense_small_kernel<<<GG, 64, 0, stream>>>(oAng, P.as_W, P.as_b, oZen, 2, 2, 1);

  dense_small_kernel<<<GG, 64, 0, stream>>>(zb2, P.sig_W0, P.sig_b0, hbA, 512, 64, 0);
  dense_small_kernel<<<GG, 64, 0, stream>>>(hbA, P.sig_W1, P.sig_b1, hbB, 64, 64, 0);
  dense_small_kernel<<<GG, 64, 0, stream>>>(hbB, P.sig_W2, P.sig_b2, oSig, 64, 2, 2);

  finalize_kernel<<<1, GG, 0, stream>>>(oLog, oZen, oSig, (float*)d_out, GG);
}